// GraphEmbedderGATNE_13786845020224
// MI455X (gfx1250) — compile-verified
//
#include <hip/hip_runtime.h>
#include <math.h>

// ---------------------------------------------------------------------------
// GATNE edge-embedding pipeline for MI455X (gfx1250, wave32, WMMA).
//
// N=200000 entities, R=4, D=128, U=32, A=16, E=131072 edges, K=10 neighbors.
//
//   * gather-bandwidth bound (~1.6 GB of u/emb gathers) -> coalesced b128 loads
//   * M staged in LDS row-pair interleaved + padded: B frag = 1 conflict-free
//     ds_load_b64; W [8KB], w [256B] also in LDS; shared by 8 waves
//   * P rows padded to 132 floats so A-frag ds_load_b64 is conflict-free
//   * out[16,128] = P[16,128] x Mflat[128,128] via V_WMMA_F32_16X16X4_F32
//   * wave handles 16 edges; block = 8 waves = 128 edges; grid = 2*1024
// ---------------------------------------------------------------------------

#define EDGES      131072
#define KNBR       10
#define RREL       4
#define UDIM       32
#define ADIM       16
#define DDIM       128
#define RU         128                    // RREL*UDIM == K of the big GEMM
#define WAVE_EDGES 16
#define WAVES_PB   8
#define BLOCK_EDGES (WAVE_EDGES*WAVES_PB)   // 128
#define BLOCKS_PER_SIDE (EDGES/BLOCK_EDGES) // 1024

// ---- LDS partition (in floats) --------------------------------------------
// M pair-interleaved: sMp[pair][d][phase], pair = c>>1 (64), phase = c&1.
// Pair stride 288 (=256+32): 288 % 64banks == 32 -> the two half-waves of a
// B-frag b64 load cover disjoint 32-bank halves (conflict-free).
#define MP_STRIDE 288
#define SMP_CNT  (64*MP_STRIDE)           // 18432
#define SW_CNT   (RREL*UDIM*ADIM)         // 2048   W[r][u][a]
#define Sw_CNT   (RREL*ADIM)              // 64     w[r][a]
// P/u_r row stride 132 (=128+4): 132 % 64banks == 4 -> A-frag b64 loads put
// lane m on banks {4m+c0, +1} (lanes 0-15: ≡{0,1} mod 4; 16-31: ≡{2,3}).
#define UP_ROW   132
#define WAVE_STRIDE (WAVE_EDGES*UP_ROW + WAVE_EDGES*RREL + WAVE_EDGES) // 2112+64+16
#define SMEM_FLOATS (SMP_CNT + SW_CNT + Sw_CNT + WAVES_PB*WAVE_STRIDE)
#define SMEM_BYTES  (SMEM_FLOATS*4)       // 152320 B (2 blocks/WGP under 320KB)

typedef float v2f __attribute__((ext_vector_type(2)));
typedef float v8f __attribute__((ext_vector_type(8)));

__global__ __launch_bounds__(256)
void gatne_edge_kernel(const int*   __restrict__ edge_index, // [2,E]
                       const float* __restrict__ edge_attr,  // [E,R]
                       const int*   __restrict__ nbr,        // [2,E,K]
                       const float* __restrict__ emb,        // [N,D]
                       const float* __restrict__ uTab,       // [N,R,U] (=128 contiguous)
                       const float* __restrict__ Wg,         // [R,U,A]
                       const float* __restrict__ wg,         // [R,A]
                       const float* __restrict__ Mg,         // [R,U,D] == Mflat[128][128]
                       float*       __restrict__ out)        // [2,E,D]
{
    extern __shared__ float smem[];
    float* sMp = smem;                 // pair-interleaved M (see above)
    float* sW  = sMp + SMP_CNT;        // W[r][u][a]
    float* swv = sW + SW_CNT;          // w[r][a]

    const int tid  = threadIdx.x;
    const int lane = tid & 31;
    const int wave = tid >> 5;

    // ---- cooperative stage of M (interleave on the fly) / W / w -----------
    {
        const float4* Ms = (const float4*)Mg;
        #pragma unroll 4
        for (int i4 = tid; i4 < (RU*DDIM)/4; i4 += 256) {
            const float4 v = Ms[i4];
            const int c  = i4 >> 5;           // row c = r*32+u  (32 float4 per row)
            const int d0 = (i4 & 31) * 4;
            float* dst = sMp + (c >> 1) * MP_STRIDE + d0 * 2 + (c & 1);
            dst[0] = v.x; dst[2] = v.y; dst[4] = v.z; dst[6] = v.w;
        }
        const float4* Ws = (const float4*)Wg;  float4* Wd = (float4*)sW;
        for (int i = tid; i < SW_CNT/4; i += 256) Wd[i] = Ws[i];
        const float4* ws = (const float4*)wg;  float4* wd = (float4*)swv;
        for (int i = tid; i < Sw_CNT/4; i += 256) wd[i] = ws[i];
    }

    const int side  = blockIdx.x / BLOCKS_PER_SIDE;          // 0 = source, 1 = target
    const int edge0 = (blockIdx.x % BLOCKS_PER_SIDE) * BLOCK_EDGES + wave * WAVE_EDGES;

    float* wPU  = smem + SMP_CNT + SW_CNT + Sw_CNT + wave * WAVE_STRIDE; // [16][132]
    float* eaS  = wPU + WAVE_EDGES*UP_ROW;                               // [16][4]
    int*   entS = (int*)(eaS + WAVE_EDGES*RREL);                         // [16]

    // ---- per-wave edge metadata (lanes 0..15), prefetch emb rows ----------
    if (lane < WAVE_EDGES) {
        const int edge = edge0 + lane;
        ((float4*)eaS)[lane] = ((const float4*)edge_attr)[edge];
        const int ent = edge_index[(size_t)side*EDGES + edge];
        entS[lane] = ent;
        const char* p = (const char*)(emb + (size_t)ent*DDIM);
        __builtin_prefetch(p,       0, 0);   // global_prefetch_b8 (512B row)
        __builtin_prefetch(p + 256, 0, 0);
    }

    // ---- Phase A: u_r = mean_k u[neighbors]  ->  wPU[e][r*32+u] -----------
    // each lane owns a 16B slice of the contiguous 512B u-row: full coalescing
    for (int e = 0; e < WAVE_EDGES; ++e) {
        const int edge = edge0 + e;
        int nb = 0;
        if (lane < KNBR) nb = nbr[((size_t)side*EDGES + edge)*KNBR + lane];
        float ax = 0.f, ay = 0.f, az = 0.f, aw = 0.f;
        #pragma unroll
        for (int kk = 0; kk < KNBR; ++kk) {
            const int idx = __shfl(nb, kk, 32);
            const float4 t = ((const float4*)(uTab + (size_t)idx*RU))[lane];
            ax += t.x; ay += t.y; az += t.z; aw += t.w;
        }
        const float s = 1.0f / (float)KNBR;
        float4 o; o.x = ax*s; o.y = ay*s; o.z = az*s; o.w = aw*s;
        ((float4*)(wPU + e*UP_ROW))[lane] = o;
    }

    __syncthreads();   // sMp/sW/swv ready (overlapped with the gather phase)

    // ---- Phase B: attention scores -> softmax -> v -> P (in place) --------
    for (int e = 0; e < WAVE_EDGES; ++e) {
        const float ea0 = eaS[e*4+0], ea1 = eaS[e*4+1],
                    ea2 = eaS[e*4+2], ea3 = eaS[e*4+3];
        const int a  = lane & 15;
        const int r1 = lane >> 4;                 // lane<16 -> r=0 ; else r=1 (and r+2)
        const float* pu = wPU + e*UP_ROW;
        float h1 = 0.f, h2 = 0.f;
        #pragma unroll 4
        for (int uu = 0; uu < UDIM; ++uu) {
            const float wr = ea0*sW[(0*UDIM+uu)*ADIM + a] + ea1*sW[(1*UDIM+uu)*ADIM + a]
                           + ea2*sW[(2*UDIM+uu)*ADIM + a] + ea3*sW[(3*UDIM+uu)*ADIM + a];
            h1 = fmaf(pu[ r1     *UDIM + uu], wr, h1);
            h2 = fmaf(pu[(r1 + 2)*UDIM + uu], wr, h2);
        }
        const float wra = ea0*swv[a] + ea1*swv[ADIM+a] + ea2*swv[2*ADIM+a] + ea3*swv[3*ADIM+a];
        float t1 = tanhf(h1) * wra;
        float t2 = tanhf(h2) * wra;
        #pragma unroll
        for (int mk = 8; mk >= 1; mk >>= 1) {     // reduce over a within 16-lane halves
            t1 += __shfl_xor(t1, mk, 32);
            t2 += __shfl_xor(t2, mk, 32);
        }
        const float s0 = __shfl(t1, 0, 32), s1 = __shfl(t1, 16, 32);
        const float s2 = __shfl(t2, 0, 32), s3 = __shfl(t2, 16, 32);
        const float mx = fmaxf(fmaxf(s0, s1), fmaxf(s2, s3));
        float x0 = __expf(s0-mx), x1 = __expf(s1-mx), x2 = __expf(s2-mx), x3 = __expf(s3-mx);
        const float inv = 1.0f / (x0 + x1 + x2 + x3);
        x0 *= inv; x1 *= inv; x2 *= inv; x3 *= inv;
        // v[u=lane] then P[e][r*32+u] = ea[r]*v[u]; per-wave DS ops are in-order,
        // and each lane rewrites exactly the 4 floats it just read -> safe in place.
        float* p = wPU + e*UP_ROW;
        const float v = p[lane]*x0 + p[UDIM+lane]*x1 + p[2*UDIM+lane]*x2 + p[3*UDIM+lane]*x3;
        p[           lane] = ea0 * v;
        p[  UDIM   + lane] = ea1 * v;
        p[2*UDIM   + lane] = ea2 * v;
        p[3*UDIM   + lane] = ea3 * v;
    }

    // ---- Phase C: out[16,128] = P[16,128] x Mflat[128,128] via f32 WMMA ---
    // A 16x4 f32 frag: lane(l&15)=row M, VGPR0 K = 2*(l>>4), VGPR1 K+1
    //   -> one b64 from padded wPU row (conflict-free).
    // B 4x16 f32 frag: rows (c0, c0+1) at col = t*16+(l&15)
    //   -> one b64 from pair-interleaved sMp (conflict-free).
    const int m  = lane & 15;
    const int hh = lane >> 4;
    v8f acc[8] = {};
    const float* pA = wPU + m*UP_ROW;
    #pragma unroll 8
    for (int k = 0; k < 32; ++k) {
        const v2f af = *(const v2f*)(pA + 4*k + 2*hh);
        const float* pB = sMp + (2*k + hh) * MP_STRIDE + 2*m;
        v2f bfr[8];
        #pragma unroll
        for (int t = 0; t < 8; ++t) bfr[t] = *(const v2f*)(pB + t*32);
        #pragma unroll
        for (int t = 0; t < 8; ++t)
            acc[t] = __builtin_amdgcn_wmma_f32_16x16x4_f32(
                false, af, false, bfr[t], (short)0, acc[t], false, false);
    }

    // ---- Phase D: + emb gather, L2 normalize over D, store ----------------
    // C frag: lanes 0-15 hold rows 0..7 in vgpr j, lanes 16-31 rows 8..15; col = t*16+m.
    #pragma unroll
    for (int j = 0; j < 8; ++j) {
        const int row  = hh ? (j + 8) : j;
        const int edge = edge0 + row;
        const int ent  = entS[row];
        const float* brow = emb + (size_t)ent*DDIM + m;
        float vals[8];
        float ss = 0.f;
        #pragma unroll
        for (int t = 0; t < 8; ++t) {
            const float x = acc[t][j] + brow[t*16];
            vals[t] = x;
            ss = fmaf(x, x, ss);
        }
        #pragma unroll
        for (int mk = 8; mk >= 1; mk >>= 1) ss += __shfl_xor(ss, mk, 32);
        const float invn = 1.0f / fmaxf(sqrtf(ss), 1e-12f);
        float* orow = out + ((size_t)side*EDGES + edge)*DDIM + m;
        #pragma unroll
        for (int t = 0; t < 8; ++t) orow[t*16] = vals[t] * invn;
    }
}

extern "C" void kernel_launch(void* const* d_in, const int* in_sizes, int n_in,
                              void* d_out, int out_size, void* d_ws, size_t ws_size,
                              hipStream_t stream) {
    const int*   edge_index = (const int*)  d_in[0];
    const float* edge_attr  = (const float*)d_in[1];
    const int*   nbr        = (const int*)  d_in[2];
    const float* emb        = (const float*)d_in[3];
    const float* uTab       = (const float*)d_in[4];
    const float* Wg         = (const float*)d_in[5];
    const float* wg         = (const float*)d_in[6];
    const float* Mg         = (const float*)d_in[7];
    float*       out        = (float*)d_out;

    (void)in_sizes; (void)n_in; (void)out_size; (void)d_ws; (void)ws_size;

    // ~149KB dynamic LDS per block (gfx1250 WGP has 320KB -> 2 blocks/WGP)
    hipFuncSetAttribute((const void*)gatne_edge_kernel,
                        hipFuncAttributeMaxDynamicSharedMemorySize, SMEM_BYTES);

    dim3 grid(2 * BLOCKS_PER_SIDE);   // 2048 blocks: side-major
    dim3 block(256);                  // 8 waves (wave32)
    gatne_edge_kernel<<<grid, block, SMEM_BYTES, stream>>>(
        edge_index, edge_attr, nbr, emb, uTab, Wg, wg, Mg, out);
}